// LlamaAttention_75600014344308
// MI455X (gfx1250) — compile-verified
//
#include <hip/hip_runtime.h>
#include <hip/hip_bf16.h>
#include <math.h>

// ---- problem constants (match reference) ----
#define BB   2
#define SS   2048
#define DD   2048
#define NH   16
#define NKV  4
#define HD   128
#define QK_SCALE 0.08838834764831845f  // 1/sqrt(128)
#define NEGBIG  -1.0e9f

typedef float v2f __attribute__((ext_vector_type(2)));
typedef float v8f __attribute__((ext_vector_type(8)));

// D = A(16x4) * B(4x16) + C, full fp32 WMMA
__device__ __forceinline__ v8f wmma4(v2f a, v2f b, v8f c) {
    return __builtin_amdgcn_wmma_f32_16x16x4_f32(
        /*neg_a=*/false, a, /*neg_b=*/false, b,
        /*c_mod=*/(short)0, c, /*reuse_a=*/false, /*reuse_b=*/false);
}

// ---------------------------------------------------------------------------
// fp32 WMMA GEMM: C[M,N] = X[M,K] @ W[K,N], row-major.
// One wave computes a 64x64 output tile: 4 (M) x 4 (N) accumulators, so each
// K-chunk issues 16 WMMAs against 4 A-fragments + 4 B-fragments (12 loads).
// M multiple of 64, N multiple of 64, K multiple of 4 (true here).
// ---------------------------------------------------------------------------
__global__ __launch_bounds__(32)
void gemm_f32_wmma(const float* __restrict__ X, const float* __restrict__ W,
                   float* __restrict__ C, int M, int N, int K) {
    const int lane = threadIdx.x;
    const int h = lane >> 4;       // half-wave: 0 or 1
    const int r = lane & 15;       // row/col index within 16-tile
    const int n0 = blockIdx.x * 64;
    const int m0 = blockIdx.y * 64;

    v8f acc[4][4] = {};            // [m-subtile][n-subtile] -> 128 VGPRs
    const float* xrow[4];
#pragma unroll
    for (int mm = 0; mm < 4; ++mm)
        xrow[mm] = X + (size_t)(m0 + 16 * mm + r) * K;

    for (int k = 0; k < K; k += 4) {
        // A fragments: lane(h,r) holds X[row][k+2h .. k+2h+1] (8B loads)
        v2f a[4];
#pragma unroll
        for (int mm = 0; mm < 4; ++mm) {
            float2 av = *(const float2*)(xrow[mm] + k + 2 * h);
            a[mm].x = av.x; a[mm].y = av.y;
        }
        // B fragments: lane(h,r) holds W[k+2h+v][n0+16t+r]
        const float* wb = W + (size_t)(k + 2 * h) * N + n0 + r;
        v2f b[4];
#pragma unroll
        for (int t = 0; t < 4; ++t) {
            b[t].x = wb[16 * t];
            b[t].y = wb[16 * t + N];
        }
#pragma unroll
        for (int mm = 0; mm < 4; ++mm)
#pragma unroll
            for (int t = 0; t < 4; ++t)
                acc[mm][t] = wmma4(a[mm], b[t], acc[mm][t]);
    }

#pragma unroll
    for (int mm = 0; mm < 4; ++mm)
#pragma unroll
        for (int t = 0; t < 4; ++t)
#pragma unroll
            for (int j = 0; j < 8; ++j)
                C[(size_t)(m0 + 16 * mm + j + 8 * h) * N + n0 + 16 * t + r] =
                    acc[mm][t][j];
}

// ---------------------------------------------------------------------------
// RoPE, in place on a (B, S, H, HD) fp32 buffer. One thread per (.., pair i).
// out[2i] = x[2i]*cos - x[2i+1]*sin ; out[2i+1] = x[2i]*sin + x[2i+1]*cos
// angle = s * 10000^(-i/64)
// ---------------------------------------------------------------------------
__global__ __launch_bounds__(256)
void rope_kernel(float* __restrict__ x, int H, int total_pairs) {
    int idx = blockIdx.x * blockDim.x + threadIdx.x;
    if (idx >= total_pairs) return;
    const int i = idx & 63;          // pair index within head dim
    const int row = idx >> 6;        // (b*S+s)*H + h
    const int s = (row / H) % SS;
    const float LN1E4_OVER_64 = 0.14391156831212787f; // ln(10000)/64
    float inv = expf(-(float)i * LN1E4_OVER_64);
    float ang = (float)s * inv;
    float c = cosf(ang), sn = sinf(ang);
    float* p = x + (size_t)row * HD + 2 * i;
    float x1 = p[0], x2 = p[1];
    p[0] = x1 * c - x2 * sn;
    p[1] = x1 * sn + x2 * c;
}

// ---------------------------------------------------------------------------
// Flash attention (causal, GQA). One wave per 16-row Q tile per (b, head).
// Q: (B,S,NH,HD)  K,V: (B,S,NKV,HD)  O: (B,S,NH*HD)
// ---------------------------------------------------------------------------
__global__ __launch_bounds__(32)
void attn_kernel(const float* __restrict__ Q, const float* __restrict__ Kb,
                 const float* __restrict__ Vb, float* __restrict__ O) {
    __shared__ float pbuf[16][17];       // P tile, padded (C-layout -> A-layout)

    const int lane = threadIdx.x;
    const int h2 = lane >> 4;
    const int r  = lane & 15;

    const int QT = SS / 16;
    int tile = blockIdx.x;
    const int qt   = tile % QT;
    const int head = (tile / QT) % NH;
    const int b    = tile / (QT * NH);
    const int kvh  = head / (NH / NKV);
    const int q0   = qt * 16;

    // ---- load Q tile (16x128) into registers, A-layout, pre-scaled ----
    v2f qa[32];
    {
        const float* qrow = Q + (((size_t)(b * SS + q0 + r)) * NH + head) * HD;
#pragma unroll
        for (int c = 0; c < 32; ++c) {
            float2 t = *(const float2*)(qrow + 4 * c + 2 * h2);
            qa[c].x = t.x * QK_SCALE;
            qa[c].y = t.y * QK_SCALE;
        }
    }

    v8f o[8] = {};
    float m[8], l[8];
#pragma unroll
    for (int j = 0; j < 8; ++j) { m[j] = -3.0e38f; l[j] = 0.0f; }

    const int ntiles = qt + 1;           // causal: kv tiles 0..qt
    for (int kt = 0; kt < ntiles; ++kt) {
        const int kv0 = kt * 16;
        const float* krow = Kb + (((size_t)(b * SS + kv0 + r)) * NKV + kvh) * HD;
        if (kt + 1 < ntiles)             // speculative prefetch of next K row
            __builtin_prefetch(krow + (size_t)16 * NKV * HD, 0, 0);

        // ---- scores tile S = Q . K^T (16x16), 32 chained WMMAs over HD ----
        v8f sc = {};
#pragma unroll
        for (int c = 0; c < 32; ++c) {
            float2 t = *(const float2*)(krow + 4 * c + 2 * h2);
            v2f bv; bv.x = t.x; bv.y = t.y;
            sc = wmma4(qa[c], bv, sc);
        }

        // ---- causal mask, only the diagonal tile is partial ----
        if (kv0 + 15 > q0) {
            const int col = kv0 + r;
#pragma unroll
            for (int j = 0; j < 8; ++j) {
                int row = q0 + j + 8 * h2;
                if (col > row) sc[j] = NEGBIG;
            }
        }

        // ---- online softmax update ----
        float p[8], alpha[8];
#pragma unroll
        for (int j = 0; j < 8; ++j) {
            float v = sc[j];
            v = fmaxf(v, __shfl_xor(v, 1));
            v = fmaxf(v, __shfl_xor(v, 2));
            v = fmaxf(v, __shfl_xor(v, 4));
            v = fmaxf(v, __shfl_xor(v, 8));      // row max (within 16-lane half)
            float mn = fmaxf(m[j], v);
            alpha[j] = __expf(m[j] - mn);
            m[j] = mn;
            float pe = __expf(sc[j] - mn);
            p[j] = pe;
            float rs = pe;
            rs += __shfl_xor(rs, 1);
            rs += __shfl_xor(rs, 2);
            rs += __shfl_xor(rs, 4);
            rs += __shfl_xor(rs, 8);             // row sum
            l[j] = l[j] * alpha[j] + rs;
        }
#pragma unroll
        for (int t = 0; t < 8; ++t)
#pragma unroll
            for (int j = 0; j < 8; ++j) o[t][j] = o[t][j] * alpha[j];

        // ---- transpose P: C-layout -> A-layout through LDS ----
        __syncthreads();                 // single-wave WG: just a DS wait
#pragma unroll
        for (int j = 0; j < 8; ++j) pbuf[j + 8 * h2][r] = p[j];
        __syncthreads();
        v2f pa[4];
#pragma unroll
        for (int c = 0; c < 4; ++c) {
            pa[c].x = pbuf[r][4 * c + 2 * h2];
            pa[c].y = pbuf[r][4 * c + 2 * h2 + 1];
        }

        // ---- O += P (16x16) @ V (16x128), 8 accumulators across HD ----
#pragma unroll
        for (int c = 0; c < 4; ++c) {
            const float* vr0 = Vb + (((size_t)(b * SS + kv0 + 4 * c + 2 * h2)) * NKV + kvh) * HD;
            const float* vr1 = vr0 + (size_t)NKV * HD;
#pragma unroll
            for (int t = 0; t < 8; ++t) {
                v2f bv;
                bv.x = vr0[16 * t + r];
                bv.y = vr1[16 * t + r];
                o[t] = wmma4(pa[c], bv, o[t]);
            }
        }
    }

    // ---- normalize and write (b, s, head*HD + d) ----
#pragma unroll
    for (int j = 0; j < 8; ++j) l[j] = 1.0f / l[j];
#pragma unroll
    for (int t = 0; t < 8; ++t)
#pragma unroll
        for (int j = 0; j < 8; ++j) {
            int row = q0 + j + 8 * h2;
            O[((size_t)(b * SS + row)) * (NH * HD) + head * HD + 16 * t + r] =
                o[t][j] * l[j];
        }
}

// ---------------------------------------------------------------------------
// inputs: x, mask(unused - causal computed in-kernel), Wq, Wk, Wv, Wo
// ---------------------------------------------------------------------------
extern "C" void kernel_launch(void* const* d_in, const int* in_sizes, int n_in,
                              void* d_out, int out_size, void* d_ws, size_t ws_size,
                              hipStream_t stream) {
    (void)in_sizes; (void)n_in; (void)out_size; (void)ws_size;
    const float* x  = (const float*)d_in[0];
    const float* Wq = (const float*)d_in[2];
    const float* Wk = (const float*)d_in[3];
    const float* Wv = (const float*)d_in[4];
    const float* Wo = (const float*)d_in[5];
    float* out = (float*)d_out;

    float* ws = (float*)d_ws;
    const size_t qElems  = (size_t)BB * SS * NH * HD;   // 8,388,608
    const size_t kvElems = (size_t)BB * SS * NKV * HD;  // 2,097,152
    float* Qb = ws;
    float* Kb = Qb + qElems;
    float* Vb = Kb + kvElems;
    float* AO = Vb + kvElems;                           // total ~84 MB

    const int M = BB * SS;  // 4096
    dim3 wave(32);

    // QKV projections (64x64 tile per wave)
    gemm_f32_wmma<<<dim3((NH  * HD) / 64, M / 64), wave, 0, stream>>>(x, Wq, Qb, M, NH  * HD, DD);
    gemm_f32_wmma<<<dim3((NKV * HD) / 64, M / 64), wave, 0, stream>>>(x, Wk, Kb, M, NKV * HD, DD);
    gemm_f32_wmma<<<dim3((NKV * HD) / 64, M / 64), wave, 0, stream>>>(x, Wv, Vb, M, NKV * HD, DD);

    // RoPE on Q and K
    int qp = BB * SS * NH  * (HD / 2);
    int kp = BB * SS * NKV * (HD / 2);
    rope_kernel<<<(qp + 255) / 256, 256, 0, stream>>>(Qb, NH,  qp);
    rope_kernel<<<(kp + 255) / 256, 256, 0, stream>>>(Kb, NKV, kp);

    // causal flash attention with GQA
    attn_kernel<<<BB * NH * (SS / 16), wave, 0, stream>>>(Qb, Kb, Vb, AO);

    // output projection
    gemm_f32_wmma<<<dim3(DD / 64, M / 64), wave, 0, stream>>>(AO, Wo, out, M, DD, DD);
}